// VectorizationLoss_87256555586163
// MI455X (gfx1250) — compile-verified
//
#include <hip/hip_runtime.h>
#include <hip/hip_bf16.h>

typedef __attribute__((ext_vector_type(2))) float v2f;
typedef __attribute__((ext_vector_type(8))) float v8f;

#define BIGF 3.0e38f

// ---------------------------------------------------------------------------
// Kernel 1: chamfer distance via V_WMMA_F32_16X16X4_F32.
// One wave32 per (b,s,t) pair. out[b,s] is a 128x2 matrix A, target[b,t] is
// 128x2 -> B^T. xy = A * B^T computed in 16x16 tiles with K=4 (D=2 padded).
// dist[b,s,t] = mean_n min_m D2 + mean_m min_n D2, D2 = max(xx+yy-2xy, 0).
// ---------------------------------------------------------------------------
__global__ __launch_bounds__(256) void chamfer_wmma_kernel(
    const float* __restrict__ outp,     // (16,32,128,2)
    const float* __restrict__ target,   // (16,32,128,2)
    float* __restrict__ dist_ws)        // (16,32,32)
{
    const int wave = threadIdx.x >> 5;
    const int lane = threadIdx.x & 31;
    const int l16  = lane & 15;
    const int half = lane >> 4;

    const int pair = blockIdx.x * 8 + wave;      // grid sized exactly: 16384 pairs
    const int b = pair >> 10;
    const int s = (pair >> 5) & 31;
    const int t = pair & 31;

    const float2* O = reinterpret_cast<const float2*>(outp)   + (b * 32 + s) * 128;
    const float2* T = reinterpret_cast<const float2*>(target) + (b * 32 + t) * 128;

    // colMin accumulators: lane owns column tj*16 + l16 (per half; halves merged later)
    float cmin8[8];
#pragma unroll
    for (int j = 0; j < 8; ++j) cmin8[j] = BIGF;

    float rowSum = 0.0f;

    for (int ti = 0; ti < 8; ++ti) {
        // A tile: 16x4 f32. Lanes 0-15: VGPR0=K0(x), VGPR1=K1(y). Lanes 16-31 (K2,K3)=0.
        v2f a;
        if (half == 0) {
            float2 o = O[ti * 16 + l16];
            a.x = o.x; a.y = o.y;
        } else {
            a.x = 0.0f; a.y = 0.0f;
        }
        // xx for the 8 rows this lane's C-VGPRs cover: row = ti*16 + 8*half + r
        float xxsq = a.x * a.x + a.y * a.y;     // valid in lanes 0-15
        float xx8[8];
#pragma unroll
        for (int r = 0; r < 8; ++r) xx8[r] = __shfl(xxsq, half * 8 + r, 32);

        float rmin[8];
#pragma unroll
        for (int r = 0; r < 8; ++r) rmin[r] = BIGF;

#pragma unroll
        for (int tj = 0; tj < 8; ++tj) {
            // B tile: 4x16 f32, symmetric layout to A.
            v2f bb;
            if (half == 0) {
                float2 tg = T[tj * 16 + l16];
                bb.x = tg.x; bb.y = tg.y;
            } else {
                bb.x = 0.0f; bb.y = 0.0f;
            }
            float yysq = bb.x * bb.x + bb.y * bb.y; // valid in lanes 0-15
            float yyv  = __shfl(yysq, l16, 32);     // col = tj*16 + l16 in both halves

            v8f c = {0.f, 0.f, 0.f, 0.f, 0.f, 0.f, 0.f, 0.f};
            c = __builtin_amdgcn_wmma_f32_16x16x4_f32(
                    false, a, false, bb, (short)0, c, false, false);

#pragma unroll
            for (int r = 0; r < 8; ++r) {
                float d = fmaxf(xx8[r] + yyv - 2.0f * c[r], 0.0f);
                rmin[r]   = fminf(rmin[r], d);
                cmin8[tj] = fminf(cmin8[tj], d);
            }
        }
        // Row mins: reduce across the 16 lanes of each half (rows ti*16+8*half+r).
#pragma unroll
        for (int r = 0; r < 8; ++r) {
            float v = rmin[r];
            v = fminf(v, __shfl_xor(v, 1, 32));
            v = fminf(v, __shfl_xor(v, 2, 32));
            v = fminf(v, __shfl_xor(v, 4, 32));
            v = fminf(v, __shfl_xor(v, 8, 32));
            rowSum += v;   // replicated within each half
        }
    }

    // rows: low half holds sum for rows {..+0..7}, high half for {..+8..15}
    float rowTotal = rowSum + __shfl_xor(rowSum, 16, 32);

    // cols: merge half-row mins, then sum (each column counted twice -> *0.5)
    float colSum = 0.0f;
#pragma unroll
    for (int tj = 0; tj < 8; ++tj) {
        float v = fminf(cmin8[tj], __shfl_xor(cmin8[tj], 16, 32));
        colSum += v;
    }
#pragma unroll
    for (int m = 1; m < 32; m <<= 1) colSum += __shfl_xor(colSum, m, 32);
    colSum *= 0.5f;

    if (lane == 0)
        dist_ws[pair] = (rowTotal + colSum) * (1.0f / 128.0f);
}

// ---------------------------------------------------------------------------
// Kernel 2: control-point cost cp[b,s,t] (P=2, CP_WEIGHT=1).
// ---------------------------------------------------------------------------
__device__ __forceinline__ float cpc2(float ax, float ay, float bx, float by) {
    float dx = ax - bx, dy = ay - by;
    return 0.5f * (dx * dx + dy * dy);     // sum_d (a-b)^2 / P, P=2
}

__global__ void cp_cost_kernel(const float* __restrict__ target,      // (16,32,128,2)
                               const float* __restrict__ ctrl,        // (16,32,4,2)
                               float* __restrict__ cp_ws)              // (16,32,32)
{
    int idx = blockIdx.x * blockDim.x + threadIdx.x;
    if (idx >= 16 * 32 * 32) return;
    int b = idx >> 10, s = (idx >> 5) & 31, t = idx & 31;

    const float* cps = ctrl + ((b * 32 + s) * 4) * 2;
    float fx = cps[0], fy = cps[1];   // controlpoints[...,0,:]
    float lx = cps[6], ly = cps[7];   // controlpoints[...,3,:]

    const float* tg = target + ((b * 32 + t) * 128) * 2;
    float t0x = tg[0],   t0y = tg[1];
    float tLx = tg[254], tLy = tg[255];

    float zz = cpc2(fx, fy, t0x, t0y);
    float zo = cpc2(fx, fy, tLx, tLy);
    float oz = cpc2(lx, ly, t0x, t0y);
    float oo = cpc2(lx, ly, tLx, tLy);

    cp_ws[idx] = fminf(zz + oo, zo + oz);
}

// ---------------------------------------------------------------------------
// Kernel 3: Hungarian (Jonker-Volgenant shortest augmenting path) on the 16
// 32x32 cp matrices; one thread per batch. Writes [dist_losses | cp_losses].
// ---------------------------------------------------------------------------
__global__ void hungarian_kernel(const float* __restrict__ dist_ws,
                                 const float* __restrict__ cp_ws,
                                 float* __restrict__ outp)
{
    int bi = blockIdx.x * blockDim.x + threadIdx.x;
    if (bi >= 16) return;

    const float* C  = cp_ws   + bi * 1024;   // (32,32) row s, col t
    const float* Dm = dist_ws + bi * 1024;
    const int n = 32;

    float u[33], v[33], minv[33];
    int   p[33], way[33];
    bool  used[33];

    for (int j = 0; j <= n; ++j) { u[j] = 0.f; v[j] = 0.f; p[j] = 0; way[j] = 0; }

    for (int i = 1; i <= n; ++i) {
        p[0] = i;
        int j0 = 0;
        for (int j = 0; j <= n; ++j) { minv[j] = BIGF; used[j] = false; }
        do {
            used[j0] = true;
            int i0 = p[j0], j1 = 0;
            float delta = BIGF;
            for (int j = 1; j <= n; ++j) {
                if (!used[j]) {
                    float cur = C[(i0 - 1) * n + (j - 1)] - u[i0] - v[j];
                    if (cur < minv[j]) { minv[j] = cur; way[j] = j0; }
                    if (minv[j] < delta) { delta = minv[j]; j1 = j; }
                }
            }
            for (int j = 0; j <= n; ++j) {
                if (used[j]) { u[p[j]] += delta; v[j] -= delta; }
                else           minv[j] -= delta;
            }
            j0 = j1;
        } while (p[j0] != 0);
        do { int j1 = way[j0]; p[j0] = p[j1]; j0 = j1; } while (j0);
    }

    float dsum = 0.f, csum = 0.f;
    for (int j = 1; j <= n; ++j) {
        int r = p[j] - 1, c = j - 1;
        dsum += Dm[r * n + c];
        csum += C[r * n + c];
    }
    outp[bi]      = dsum;   // dist_losses
    outp[16 + bi] = csum;   // cp_losses
}

// ---------------------------------------------------------------------------
extern "C" void kernel_launch(void* const* d_in, const int* in_sizes, int n_in,
                              void* d_out, int out_size, void* d_ws, size_t ws_size,
                              hipStream_t stream) {
    const float* out_p = (const float*)d_in[0];   // (16,32,128,2)
    const float* tgt_p = (const float*)d_in[1];   // (16,32,128,2)
    const float* cpt_p = (const float*)d_in[2];   // (16,32,4,2)

    float* dist_ws = (float*)d_ws;        // 16384 floats
    float* cpc_ws  = dist_ws + 16384;     // 16384 floats  (128 KB total scratch)

    // 16384 pairs, 8 waves (256 threads) per block -> exactly full waves.
    chamfer_wmma_kernel<<<2048, 256, 0, stream>>>(out_p, tgt_p, dist_ws);
    cp_cost_kernel<<<64, 256, 0, stream>>>(tgt_p, cpt_p, cpc_ws);
    hungarian_kernel<<<1, 16, 0, stream>>>(dist_ws, cpc_ws, (float*)d_out);
}